// Attention_1726576854657
// MI455X (gfx1250) — compile-verified
//
#include <hip/hip_runtime.h>
#include <hip/hip_bf16.h>
#include <math.h>

// ---------------------------------------------------------------------------
// Attention forward for MI455X (gfx1250), fp32 via V_WMMA_F32_16X16X4_F32,
// with GLOBAL_LOAD_ASYNC_TO_LDS_B128 double-buffered tile staging (ASYNCcnt).
// B=2, T=2048, C=1024, H=16, D=64.
// ---------------------------------------------------------------------------

typedef float v2f __attribute__((ext_vector_type(2)));
typedef float v8f __attribute__((ext_vector_type(8)));

#define Bc 2
#define Tc 2048
#define Cc 1024
#define Hc 16
#define Dc 64
#define Mrows (Bc * Tc)        // 4096 total rows
#define N_QKV (3 * Cc)         // 3072

__device__ __forceinline__ v8f wmma_f32(v2f a, v2f b, v8f c) {
  // D = A(16x4) * B(4x16) + C(16x16), fp32, wave32
  return __builtin_amdgcn_wmma_f32_16x16x4_f32(
      /*neg_a=*/false, a, /*neg_b=*/false, b,
      /*c_mod=*/(short)0, c, /*reuse_a=*/false, /*reuse_b=*/false);
}

// Low 32 bits of a generic pointer to LDS = LDS byte offset (ISA 10.2:
// LDS aperture mapping truncates to addr[31:0]).
__device__ __forceinline__ unsigned int lds_off32(const void* p) {
  return (unsigned int)(unsigned long long)(uintptr_t)p;
}

// 16-byte async copy global -> LDS, tracked by ASYNCcnt (no VGPR round-trip).
__device__ __forceinline__ void async_b128(const void* gptr, unsigned int lds) {
  asm volatile("global_load_async_to_lds_b128 %0, %1, off"
               :: "v"(lds), "v"((unsigned long long)(uintptr_t)gptr)
               : "memory");
}

__device__ __forceinline__ void wait_async_le4() {
#if __has_builtin(__builtin_amdgcn_s_wait_asynccnt)
  __builtin_amdgcn_s_wait_asynccnt(4);
#else
  asm volatile("s_wait_asynccnt 0x4" ::: "memory");
#endif
}
__device__ __forceinline__ void wait_async_le0() {
#if __has_builtin(__builtin_amdgcn_s_wait_asynccnt)
  __builtin_amdgcn_s_wait_asynccnt(0);
#else
  asm volatile("s_wait_asynccnt 0x0" ::: "memory");
#endif
}

// ---------------------------------------------------------------------------
// Tiled GEMM: C[M x N] = A[M x K] * W[K x N]  (+bias) (* row mask)
// Block: 256 threads = 8 waves in a 2(M) x 4(N) grid; block tile 128x128;
// wave tile 64x32 = 4x2 WMMA accumulators. K fixed at 1024.
// Double-buffered LDS tiles filled by async B128 copies.
// ---------------------------------------------------------------------------
__global__ __launch_bounds__(256, 2)
void gemm128_wmma(const float* __restrict__ A, const float* __restrict__ W,
                  float* __restrict__ Cmat, int N,
                  const float* __restrict__ bias,
                  const float* __restrict__ rowmask) {
  __shared__ float xs[2][128 * 17];   // A tile 128x16, pad 1 (rows 64B-contig)
  __shared__ float ws[2][16 * 132];   // W tile 16x128, pad 4

  const int tid  = threadIdx.x;
  const int lane = tid & 31;
  const int wave = tid >> 5;
  const int wm   = wave >> 2;      // 0..1 (M dir)
  const int wn   = wave & 3;       // 0..3 (N dir)
  const int l16  = lane & 15;
  const int hi   = lane >> 4;      // half of wave

  const int bm = blockIdx.y * 128;
  const int bn = blockIdx.x * 128;
  const int K  = Cc;               // 1024
  const int NT = K / 16;           // 64 k-tiles

  v8f acc[4][2];
#pragma unroll
  for (int mt = 0; mt < 4; ++mt)
#pragma unroll
    for (int nt = 0; nt < 2; ++nt) acc[mt][nt] = (v8f)0.0f;

  // 4 wave-wide async ops per thread per tile: 2 for xs (128x16), 2 for ws.
  auto issue_tile = [&](int buf, int kb) {
    float* xsb = &xs[buf][0];
    float* wsb = &ws[buf][0];
#pragma unroll
    for (int e = 0; e < 2; ++e) {
      int idx = tid + e * 256;        // 0..511 B128 chunks
      int r = idx >> 2, q = (idx & 3) * 4;   // 128 rows x 4 chunks of 4 floats
      async_b128(&A[(size_t)(bm + r) * K + kb + q],
                 lds_off32(&xsb[r * 17 + q]));
    }
#pragma unroll
    for (int e = 0; e < 2; ++e) {
      int idx = tid + e * 256;
      int r = idx >> 5, q = (idx & 31) * 4;  // 16 rows x 32 chunks
      async_b128(&W[(size_t)(kb + r) * N + bn + q],
                 lds_off32(&wsb[r * 132 + q]));
    }
  };

  issue_tile(0, 0);
  for (int t = 0; t < NT; ++t) {
    const int cur = t & 1;
    if (t + 1 < NT) {
      issue_tile(cur ^ 1, (t + 1) * 16);  // prefetch next tile into other buf
      wait_async_le4();                   // current tile's 4 ops complete
    } else {
      wait_async_le0();
    }
    __syncthreads();                      // all waves confirmed their portion

    const float* xsb = &xs[cur][0];
    const float* wsb = &ws[cur][0];
#pragma unroll
    for (int kc = 0; kc < 4; ++kc) {      // 4 chained k=4 WMMA steps
      const int k0 = kc * 4 + hi * 2;
      v2f a[4], b[2];
#pragma unroll
      for (int mt = 0; mt < 4; ++mt) {
        int row = wm * 64 + mt * 16 + l16;    // A frag: m=lane%16, k=elem+2*hi
        a[mt].x = xsb[row * 17 + k0];
        a[mt].y = xsb[row * 17 + k0 + 1];
      }
#pragma unroll
      for (int nt = 0; nt < 2; ++nt) {
        int col = wn * 32 + nt * 16 + l16;    // B frag: n=lane%16, k=elem+2*hi
        b[nt].x = wsb[(k0 + 0) * 132 + col];
        b[nt].y = wsb[(k0 + 1) * 132 + col];
      }
#pragma unroll
      for (int mt = 0; mt < 4; ++mt)
#pragma unroll
        for (int nt = 0; nt < 2; ++nt)
          acc[mt][nt] = wmma_f32(a[mt], b[nt], acc[mt][nt]);
    }
    __syncthreads();   // all waves done with cur before it is refilled
  }

  // epilogue: C layout -> lane holds col n=lane%16, rows = elem + 8*hi
#pragma unroll
  for (int mt = 0; mt < 4; ++mt) {
#pragma unroll
    for (int nt = 0; nt < 2; ++nt) {
      int col = bn + wn * 32 + nt * 16 + l16;
      float bv = bias ? bias[col] : 0.0f;
#pragma unroll
      for (int v = 0; v < 8; ++v) {
        int row = bm + wm * 64 + mt * 16 + hi * 8 + v;
        float val = acc[mt][nt][v] + bv;
        if (rowmask) val *= rowmask[row];
        Cmat[(size_t)row * N + col] = val;
      }
    }
  }
}

// ---------------------------------------------------------------------------
// Flash attention: one block (128 threads = 4 waves) per (b, h, 64-row i-tile).
// Each wave owns 16 Q rows (in A-frag registers). Streams 64x64 K/V tiles
// through LDS via async B128 copies, online softmax in the WMMA C-register
// layout, causal + padding masks.
// ---------------------------------------------------------------------------
__global__ __launch_bounds__(128, 2)
void flash_attn_wmma(const float* __restrict__ qkv,
                     const float* __restrict__ mask,   // (B*T) floats
                     float* __restrict__ att) {
  __shared__ float kt[64 * 68];        // K tile [j][d], pad 4
  __shared__ float vt[64 * 68];        // V tile [j][d]
  __shared__ float pb[4][16 * 68];     // per-wave P tile [m][j]

  const int tid  = threadIdx.x;
  const int lane = tid & 31;
  const int wave = tid >> 5;           // 0..3 -> 16 Q rows each
  const int l16  = lane & 15;
  const int hi   = lane >> 4;

  const int b  = blockIdx.z;
  const int h  = blockIdx.y;
  const int i0 = blockIdx.x * 64;
  const float scale = 0.125f;          // D^-0.5 = 1/8

  // ---- load Q fragments for this wave's 16 rows, pre-scaled -------------
  const int qrow = i0 + wave * 16 + l16;
  const float* qbase = qkv + ((size_t)(b * Tc + qrow)) * N_QKV + h * Dc;
  v2f qf[16];
#pragma unroll
  for (int kbq = 0; kbq < 16; ++kbq) {
    int k0 = kbq * 4 + 2 * hi;
    qf[kbq].x = qbase[k0] * scale;
    qf[kbq].y = qbase[k0 + 1] * scale;
  }

  // row padding mask for this wave's rows (C-layout: elem v <-> row 8*hi+v)
  float mi[8];
#pragma unroll
  for (int v = 0; v < 8; ++v)
    mi[v] = mask[b * Tc + i0 + wave * 16 + hi * 8 + v];

  float rmax[8], rsum[8];
#pragma unroll
  for (int v = 0; v < 8; ++v) { rmax[v] = -INFINITY; rsum[v] = 0.0f; }
  v8f o[4];
#pragma unroll
  for (int dt = 0; dt < 4; ++dt) o[dt] = (v8f)0.0f;

  const int jend = i0 + 64;            // causal: only j-blocks with j <= i
  for (int j0 = 0; j0 < jend; j0 += 64) {
    // ---- async K/V tile fill (64x64 each): 16 B128 ops/thread ----------
#pragma unroll
    for (int e = 0; e < 8; ++e) {
      int idx = tid + e * 128;               // 0..1023 B128 chunks
      int r = idx >> 4, q = (idx & 15) * 4;  // 64 rows x 16 chunks
      const float* src =
          qkv + ((size_t)(b * Tc + j0 + r)) * N_QKV + h * Dc + q;
      async_b128(src + Cc,     lds_off32(&kt[r * 68 + q]));  // K at +C
      async_b128(src + 2 * Cc, lds_off32(&vt[r * 68 + q]));  // V at +2C
    }
    wait_async_le0();
    __syncthreads();

    // ---- S = Q * K^T  (16 x 64 per wave) --------------------------------
    v8f s[4];
#pragma unroll
    for (int nt = 0; nt < 4; ++nt) s[nt] = (v8f)0.0f;
#pragma unroll
    for (int kb = 0; kb < 16; ++kb) {
      int k0 = kb * 4 + 2 * hi;
#pragma unroll
      for (int nt = 0; nt < 4; ++nt) {
        int n = nt * 16 + l16;         // B[k=d][n=j] = kt[j][d]
        v2f bb;
        bb.x = kt[n * 68 + k0];
        bb.y = kt[n * 68 + k0 + 1];
        s[nt] = wmma_f32(qf[kb], bb, s[nt]);
      }
    }

    // ---- mask + online softmax (all in C-register layout) ---------------
    float mj[4];
#pragma unroll
    for (int nt = 0; nt < 4; ++nt) mj[nt] = mask[b * Tc + j0 + nt * 16 + l16];

#pragma unroll
    for (int v = 0; v < 8; ++v) {
      int irow = i0 + wave * 16 + hi * 8 + v;
      float mx = rmax[v];
#pragma unroll
      for (int nt = 0; nt < 4; ++nt) {
        int j = j0 + nt * 16 + l16;
        float sv = s[nt][v];
        bool dead = (j > irow) | (mj[nt] == 0.0f) | (mi[v] == 0.0f);
        sv = dead ? -3.0e38f : sv;
        s[nt][v] = sv;
        mx = fmaxf(mx, sv);
      }
      // row max across the 16 lanes of this half-wave
      mx = fmaxf(mx, __shfl_xor(mx, 1));
      mx = fmaxf(mx, __shfl_xor(mx, 2));
      mx = fmaxf(mx, __shfl_xor(mx, 4));
      mx = fmaxf(mx, __shfl_xor(mx, 8));

      float corr = __expf(rmax[v] - mx);
      rmax[v] = mx;
      float ls = 0.0f;
#pragma unroll
      for (int nt = 0; nt < 4; ++nt) {
        float p = __expf(s[nt][v] - mx);
        s[nt][v] = p;
        ls += p;
      }
      ls += __shfl_xor(ls, 1);
      ls += __shfl_xor(ls, 2);
      ls += __shfl_xor(ls, 4);
      ls += __shfl_xor(ls, 8);
      rsum[v] = rsum[v] * corr + ls;
#pragma unroll
      for (int dt = 0; dt < 4; ++dt) o[dt][v] *= corr;   // rescale O rows
    }

    // ---- P: C-layout -> LDS -> A-layout ---------------------------------
#pragma unroll
    for (int nt = 0; nt < 4; ++nt)
#pragma unroll
      for (int v = 0; v < 8; ++v)
        pb[wave][(hi * 8 + v) * 68 + nt * 16 + l16] = s[nt][v];
    // same-wave LDS ops are in-order: safe to read back without barrier

    // ---- O += P * V ------------------------------------------------------
#pragma unroll
    for (int kb = 0; kb < 16; ++kb) {
      int k0 = kb * 4 + 2 * hi;
      v2f a;
      a.x = pb[wave][l16 * 68 + k0];       // A frag: m=lane%16, k=elem+2*hi
      a.y = pb[wave][l16 * 68 + k0 + 1];
#pragma unroll
      for (int dt = 0; dt < 4; ++dt) {
        v2f bb;                             // B[k=j][n=d] = vt[j][d]
        bb.x = vt[(k0 + 0) * 68 + dt * 16 + l16];
        bb.y = vt[(k0 + 1) * 68 + dt * 16 + l16];
        o[dt] = wmma_f32(a, bb, o[dt]);
      }
    }
    __syncthreads();   // protect kt/vt before next j-block refill
  }

  // ---- normalize and store --------------------------------------------
#pragma unroll
  for (int dt = 0; dt < 4; ++dt) {
#pragma unroll
    for (int v = 0; v < 8; ++v) {
      int irow = i0 + wave * 16 + hi * 8 + v;
      float val = o[dt][v] / rsum[v];
      att[((size_t)(b * Tc + irow)) * Cc + h * Dc + dt * 16 + l16] = val;
    }
  }
}

// ---------------------------------------------------------------------------
extern "C" void kernel_launch(void* const* d_in, const int* in_sizes, int n_in,
                              void* d_out, int out_size, void* d_ws, size_t ws_size,
                              hipStream_t stream) {
  const float* x     = (const float*)d_in[0];   // (B,T,C)
  const float* m     = (const float*)d_in[1];   // (B,T,1)
  const float* w_qkv = (const float*)d_in[2];   // (C,3C)
  const float* w_out = (const float*)d_in[3];   // (C,C)
  const float* b_out = (const float*)d_in[4];   // (C,)
  float* out = (float*)d_out;

  float* qkv = (float*)d_ws;                          // 4096*3072 floats
  float* att = qkv + (size_t)Mrows * N_QKV;           // 4096*1024 floats

  // 1) qkv = x @ w_qkv        (4096x1024 @ 1024x3072)
  {
    dim3 grid(N_QKV / 128, Mrows / 128);              // (24, 32)
    gemm128_wmma<<<grid, 256, 0, stream>>>(x, w_qkv, qkv, N_QKV,
                                           nullptr, nullptr);
  }
  // 2) flash attention -> att (4096x1024)
  {
    dim3 grid(Tc / 64, Hc, Bc);                       // (32, 16, 2)
    flash_attn_wmma<<<grid, 128, 0, stream>>>(qkv, m, att);
  }
  // 3) out = (att @ w_out + b_out) * m
  {
    dim3 grid(Cc / 128, Mrows / 128);                 // (8, 32)
    gemm128_wmma<<<grid, 256, 0, stream>>>(att, w_out, out, Cc,
                                           b_out, m);
  }
}